// ActivationGATLayerIsotropic_83476984365549
// MI455X (gfx1250) — compile-verified
//
#include <hip/hip_runtime.h>

typedef __attribute__((ext_vector_type(2))) float v2f;
typedef __attribute__((ext_vector_type(8))) float v8f;

#define BN_EPS 1e-5f

// ---------------------------------------------------------------- zero ws
__global__ void agi_zero_kernel(float* __restrict__ p, int n) {
    int i = blockIdx.x * 256 + threadIdx.x;
    if (i < n) p[i] = 0.0f;
}

// ------------------------------------------------- edge scatter: agg[dst] += h[src]*norm[src]
// one edge per wave (32 lanes), 2 features per lane (D==64)
__global__ void agi_scatter_kernel(const float* __restrict__ h,
                                   const float* __restrict__ norm,
                                   const int*   __restrict__ src,
                                   const int*   __restrict__ dst,
                                   float*       __restrict__ agg,
                                   int E) {
    const int D = 64;
    long tid  = (long)blockIdx.x * blockDim.x + threadIdx.x;
    int  lane = (int)(tid & 31);
    long e    = tid >> 5;
    if (e >= E) return;
    int   s  = src[e];
    int   t  = dst[e];
    float nv = norm[s];
    int   d0 = lane * 2;
    v2f hv = *(const v2f*)(h + (long)s * D + d0);
    atomicAdd(agg + (long)t * D + d0,     hv.x * nv);
    atomicAdd(agg + (long)t * D + d0 + 1, hv.y * nv);
}

// ------------------------------------------------- column sum & sumsq via V_WMMA_F32_16X16X4_F32
// A = ones in row 0 only => D[0, n] = sum_k B[k, n].  Each wave owns one
// 16-feature group; each iteration folds 4 node-rows into the accumulators.
__global__ void agi_colstats_wmma_kernel(const float* __restrict__ agg,
                                         float* __restrict__ colsum,
                                         float* __restrict__ colsq,
                                         int N) {
    const int D    = 64;
    int lane = threadIdx.x & 31;
    int wave = threadIdx.x >> 5;
    int fg   = wave & 3;        // feature group 0..3 (16 features each)
    int half = wave >> 2;       // 0/1: which 256-node half of this block's span
    int base0 = blockIdx.x * 512 + half * 256;
    int feat  = fg * 16 + (lane & 15);

    // A-matrix (16x4 f32): lanes 0-15 hold M=lane, lanes 16-31 hold M=lane-16.
    // Put 1.0 only in row M=0 (lanes 0 and 16), both K slots per lane.
    v2f a;
    a.x = ((lane & 15) == 0) ? 1.0f : 0.0f;
    a.y = a.x;

    v8f cs = {};   // column sums
    v8f cq = {};   // column sums of squares

    int kofs = (lane >> 4) << 1;            // lanes 0-15 -> nodes +0,+1 ; lanes 16-31 -> +2,+3

    for (int base = base0; base < base0 + 256; base += 4) {
        int n0 = base + kofs;
        int n1 = n0 + 1;
        int n0c = n0 < (N - 1) ? n0 : (N - 1);
        int n1c = n1 < (N - 1) ? n1 : (N - 1);
        float v0 = agg[(long)n0c * D + feat];
        float v1 = agg[(long)n1c * D + feat];
        v0 = (n0 < N) ? v0 : 0.0f;
        v1 = (n1 < N) ? v1 : 0.0f;
        v2f b;  b.x  = v0;      b.y  = v1;
        v2f b2; b2.x = v0 * v0; b2.y = v1 * v1;
        cs = __builtin_amdgcn_wmma_f32_16x16x4_f32(false, a, false, b,  (short)0, cs, false, false);
        cq = __builtin_amdgcn_wmma_f32_16x16x4_f32(false, a, false, b2, (short)0, cq, false, false);
    }

    // C/D layout: row M=0 lives in VGPR0 of lanes 0..15 (N = lane)
    if (lane < 16) {
        atomicAdd(&colsum[feat], cs[0]);
        atomicAdd(&colsq[feat],  cq[0]);
    }
}

// ------------------------------------------------- finalize mean / inv-std
__global__ void agi_finalize_kernel(const float* __restrict__ colsum,
                                    const float* __restrict__ colsq,
                                    float* __restrict__ meanv,
                                    float* __restrict__ invv,
                                    int N) {
    int d = threadIdx.x;
    if (d < 64) {
        float rn = 1.0f / (float)N;
        float m  = colsum[d] * rn;
        float v  = colsq[d] * rn - m * m;   // biased variance (matches torch BN)
        meanv[d] = m;
        invv[d]  = rsqrtf(v + BN_EPS);
    }
}

// ------------------------------------------------- heads: BN affine + relu + norm, cat layout
__global__ void agi_output_kernel(const float* __restrict__ agg,
                                  const float* __restrict__ norm,
                                  const float* __restrict__ gamma,
                                  const float* __restrict__ beta,
                                  const float* __restrict__ meanv,
                                  const float* __restrict__ invv,
                                  float*       __restrict__ out,
                                  int N, int H) {
    const int D = 64;
    int idx = blockIdx.x * 256 + threadIdx.x;     // N*H*D = 12.8M fits in int
    int total = N * H * D;
    if (idx >= total) return;
    int d  = idx & (D - 1);
    int hh = (idx >> 6) & (H - 1);                // H is a power of two (4)
    int n  = idx >> 8;                            // idx / (H*D)
    float xhat = (agg[n * D + d] - meanv[d]) * invv[d];
    float v = xhat * gamma[hh * D + d] + beta[hh * D + d];
    v = fmaxf(v, 0.0f) * norm[n];
    out[idx] = v;
}

// ---------------------------------------------------------------- launcher
extern "C" void kernel_launch(void* const* d_in, const int* in_sizes, int n_in,
                              void* d_out, int out_size, void* d_ws, size_t ws_size,
                              hipStream_t stream) {
    const float* h     = (const float*)d_in[0];
    const float* e     = (const float*)d_in[1];
    const float* norm  = (const float*)d_in[2];
    const float* gamma = (const float*)d_in[3];
    const float* beta  = (const float*)d_in[4];
    const int*   src   = (const int*)d_in[5];
    const int*   dst   = (const int*)d_in[6];

    const int N = in_sizes[2];              // norm is [N,1]
    const int E = in_sizes[5];              // src is [E]
    const int D = in_sizes[0] / N;          // 64
    const int H = in_sizes[3] / D;          // 4

    // workspace layout
    float* agg    = (float*)d_ws;           // N*D
    float* colsum = agg + (size_t)N * D;    // D
    float* colsq  = colsum + D;             // D
    float* meanv  = colsq + D;              // D
    float* invv   = meanv + D;              // D

    // 1) zero accumulators (harness does not re-poison between replays)
    int zn = N * D + 4 * D;
    agi_zero_kernel<<<(zn + 255) / 256, 256, 0, stream>>>(agg, zn);

    // 2) edge scatter-add (L2-resident atomics; only index arrays hit HBM)
    long sthreads = (long)E * 32;
    agi_scatter_kernel<<<(int)((sthreads + 255) / 256), 256, 0, stream>>>(
        h, norm, src, dst, agg, E);

    // 3) column stats on the matrix pipe (V_WMMA_F32_16X16X4_F32)
    agi_colstats_wmma_kernel<<<(N + 511) / 512, 256, 0, stream>>>(agg, colsum, colsq, N);

    // 4) mean / inv-std
    agi_finalize_kernel<<<1, 64, 0, stream>>>(colsum, colsq, meanv, invv, N);

    // 5) fused BN-affine + relu + norm, coalesced [N, H*D] output
    int ototal = N * H * D;
    agi_output_kernel<<<(ototal + 255) / 256, 256, 0, stream>>>(
        agg, norm, gamma, beta, meanv, invv, (float*)d_out, N, H);

    // 6) e pass-through: the big HBM stream (~410 MB total), one D2D copy
    hipMemcpyAsync((float*)d_out + (size_t)ototal, e,
                   (size_t)E * D * sizeof(float), hipMemcpyDeviceToDevice, stream);
}